// LocallyConnectedLayer_63806034149978
// MI455X (gfx1250) — compile-verified
//
#include <hip/hip_runtime.h>

// Locally connected layer on MI455X (gfx1250, wave32, WMMA).
//
// Shapes: x (32,32,64,64) f32, weight (62,62,32,32,3,3) f32, bias (62,62,32) f32
// out (32,32,62,62) f32.  Per pixel: GEMM M=32(batch) x N=32(oc) x K=288(c*kh*kw).
//
// Strategy:
//  - 1 block (128 threads = 4 wave32) per output pixel (3844 blocks).
//  - im2col A tile (32x288) staged into LDS with +2 dword row pad (stride 290)
//    -> bank-conflict-free ds_load_b64 fragment reads.
//  - B tile (weights, contiguous 32x288 per pixel) streamed straight from
//    global as 8-byte loads; prefetched into L2/L0 while A is staged.
//  - Each wave computes one 16x16 output tile with 72 x v_wmma_f32_16x16x4_f32.

#define IH 64
#define IW 64
#define CIN 32
#define OCH 32
#define KK 3
#define OH 62
#define OW 62
#define P  (CIN * KK * KK)     // 288
#define LDS_STRIDE (P + 2)     // 290 dwords: bank-conflict-free, keeps 8B align

typedef __attribute__((ext_vector_type(2))) float v2f;
typedef __attribute__((ext_vector_type(8))) float v8f;

__global__ __launch_bounds__(128)
void lc_wmma_f32_kernel(const float* __restrict__ x,
                        const float* __restrict__ weight,
                        const float* __restrict__ bias,
                        float* __restrict__ out)
{
    __shared__ float sA[32 * LDS_STRIDE];   // 37,120 bytes

    const int pos = blockIdx.x;             // 0 .. 3843
    const int i   = pos / OW;
    const int j   = pos - i * OW;

    const float* __restrict__ Wp = weight + (size_t)pos * (OCH * P); // 32x288 block

    // Warm L2/L0 with this pixel's weight block while we build the A tile.
    // 128 threads x 288B stride covers the whole 36,864B block.
    __builtin_prefetch(Wp + (size_t)threadIdx.x * (P / 4), 0, 1);

    // ---- Stage im2col A tile into LDS: A[b][p] = x[b, c, i+kh, j+kw],
    //      p = c*9 + kh*3 + kw (matches reference patch order). 9216/128 = 72.
    for (int f = threadIdx.x; f < 32 * P; f += 128) {
        const int b  = f / P;
        const int p  = f - b * P;
        const int c  = p / 9;
        const int r9 = p - c * 9;
        const int kh = r9 / 3;
        const int kw = r9 - kh * 3;
        sA[b * LDS_STRIDE + p] =
            x[(((size_t)b * CIN + c) * IH + (i + kh)) * IW + (j + kw)];
    }
    __syncthreads();

    // ---- Wave / tile assignment (wave32).
    const int lane   = threadIdx.x & 31;
    const int wave   = threadIdx.x >> 5;    // 0..3
    const int laneLo = lane & 15;
    const int laneHi = lane >> 4;           // 0 or 1
    const int m0 = (wave & 1) * 16;         // batch tile
    const int n0 = (wave >> 1) * 16;        // oc tile

    // Fragment pointers.
    // A (16x4 f32): lane holds A[M=laneLo, K = 2*laneHi + {0,1}] -> ds_load_b64.
    const v2f* __restrict__ aPtr =
        (const v2f*)(sA + (m0 + laneLo) * LDS_STRIDE + 2 * laneHi);
    // B (4x16 f32): lane holds B[K = 2*laneHi + {0,1}, N=laneLo]
    //             = W[o = n0+laneLo, p = 4*kk + 2*laneHi + {0,1}] -> global b64.
    const v2f* __restrict__ bPtr =
        (const v2f*)(Wp + (n0 + laneLo) * P + 2 * laneHi);

    v8f acc = {0.f, 0.f, 0.f, 0.f, 0.f, 0.f, 0.f, 0.f};

    // K loop: 288 / 4 = 72 WMMA steps. Pointers advance 4 floats = 2 v2f.
#pragma unroll 4
    for (int kk = 0; kk < P / 4; ++kk) {
        v2f a = aPtr[2 * kk];
        v2f b = bPtr[2 * kk];
        acc = __builtin_amdgcn_wmma_f32_16x16x4_f32(
            /*neg_a=*/false, a, /*neg_b=*/false, b,
            /*c_mod=*/(short)0, acc, /*reuse_a=*/false, /*reuse_b=*/false);
    }

    // ---- Epilogue: C/D 16x16 f32 layout:
    //   lane<16 : vgpr r -> M = r,   N = laneLo
    //   lane>=16: vgpr r -> M = r+8, N = laneLo
    const int o    = n0 + laneLo;
    const float bv = bias[(size_t)pos * OCH + o];
#pragma unroll
    for (int r = 0; r < 8; ++r) {
        const int b = m0 + r + laneHi * 8;          // batch index
        out[((size_t)b * OCH + o) * (OH * OW) + pos] = acc[r] + bv;
    }
}

extern "C" void kernel_launch(void* const* d_in, const int* in_sizes, int n_in,
                              void* d_out, int out_size, void* d_ws, size_t ws_size,
                              hipStream_t stream)
{
    const float* x      = (const float*)d_in[0];  // 32*32*64*64
    const float* weight = (const float*)d_in[1];  // 62*62*32*288
    const float* bias   = (const float*)d_in[2];  // 62*62*32
    float* out          = (float*)d_out;          // 32*32*62*62

    (void)in_sizes; (void)n_in; (void)out_size; (void)d_ws; (void)ws_size;

    lc_wmma_f32_kernel<<<OH * OW, 128, 0, stream>>>(x, weight, bias, out);
}